// EquiConv_71554155151879
// MI455X (gfx1250) — compile-verified
//
#include <hip/hip_runtime.h>
#include <math.h>

// ---------------------------------------------------------------------------
// EquiConv fused kernel for MI455X (gfx1250), fp32 WMMA 16x16x4 path.
// Weights staged to LDS via global_load_async_to_lds_b128 (ASYNCcnt).
// ---------------------------------------------------------------------------

typedef __attribute__((ext_vector_type(2))) float v2f;
typedef __attribute__((ext_vector_type(8))) float v8f;

#if defined(__HIP_DEVICE_COMPILE__) && !__has_builtin(__builtin_amdgcn_wmma_f32_16x16x4_f32)
#error "wmma f32 16x16x4 builtin not available on this toolchain"
#endif

__device__ __forceinline__ v8f wmma4(v2f a, v2f b, v8f c) {
  // D = A(16x4) * B(4x16) + C(16x16), all fp32
  return __builtin_amdgcn_wmma_f32_16x16x4_f32(
      /*neg_a=*/false, a, /*neg_b=*/false, b,
      /*c_mod=*/(short)0, c, /*reuse_a=*/false, /*reuse_b=*/false);
}

// B fragment (4x16) from LDS-staged weight, row-major with padded stride.
// Lane layout mirrors A: lanes 0-15 hold rows k,k+1; lanes 16-31 rows k+2,k+3.
__device__ __forceinline__ v2f lds_b(const float* w, int stride, int kc, int h, int n) {
  int k = 4 * kc + 2 * h;
  v2f b;
  b.x = w[k * stride + n];
  b.y = w[(k + 1) * stride + n];
  return b;
}

// A fragment (16x4) from row-major global matrix.
__device__ __forceinline__ v2f glb_a(const float* X, int row0, int ld, int col0,
                                     int kc, int m, int h) {
  const float* p = X + (size_t)(row0 + m) * ld + col0 + 4 * kc + 2 * h;
  v2f a;
  a.x = p[0];
  a.y = p[1];
  return a;
}

// A fragment from per-wave LDS scratch (16 rows x stride 68).
__device__ __forceinline__ v2f lds_a(const float* hb, int kc, int m, int h) {
  const float* p = hb + m * 68 + 4 * kc + 2 * h;
  v2f a;
  a.x = p[0];
  a.y = p[1];
  return a;
}

__device__ __forceinline__ float silu(float x) {
  return x / (1.0f + __expf(-x));
}

// Cooperative async copy of a (rows x cols) row-major matrix into LDS with a
// padded stride; cols/stride are multiples of 4 -> b128 async transfers that
// bypass VGPRs entirely (tracked by ASYNCcnt).
__device__ __forceinline__ void stage_async(float* dst, const float* src, int rows,
                                            int cols, int stride, int tid) {
  const uint32_t dbase = (uint32_t)(uintptr_t)dst;  // LDS aperture: low 32 bits = DS offset
  int n4 = (rows * cols) >> 2;
  for (int i = tid; i < n4; i += 256) {
    int idx = i << 2;
    int r = idx / cols;
    int c = idx - r * cols;
    uint32_t ldsoff = dbase + (uint32_t)(r * stride + c) * 4u;
    uint64_t gaddr = (uint64_t)(uintptr_t)(src + idx);
    asm volatile("global_load_async_to_lds_b128 %0, %1, off"
                 :: "v"(ldsoff), "v"(gaddr) : "memory");
  }
}

// LDS float offsets (padded strides: +8 -> conflict-free B reads)
#define OFF_W11 0                    // 128 x 136
#define OFF_W12 (OFF_W11 + 128*136)  // 128 x 72
#define OFF_W13 (OFF_W12 + 128*72)   //  64 x 72
#define OFF_W14 (OFF_W13 + 64*72)    //  64 x 136
#define OFF_F0  (OFF_W14 + 64*136)   // 128 x 72
#define OFF_F1  (OFF_F0  + 128*72)   //  64 x 72
#define OFF_F2  (OFF_F1  + 64*72)    //  64 x 200
#define OFF_H   (OFF_F2  + 64*200)   // 8 waves x (16 x 68)
#define LDS_FLOATS (OFF_H + 8*16*68)

__global__ __launch_bounds__(256, 1)
void equiconv_wmma_kernel(const float* __restrict__ fea1,  // (E,320)
                          const float* __restrict__ fea2,  // (E,4)
                          const float* __restrict__ feaw,  // (E,128)
                          const float* __restrict__ w11, const float* __restrict__ w21,
                          const float* __restrict__ w12, const float* __restrict__ w22,
                          const float* __restrict__ w13, const float* __restrict__ w23,
                          const float* __restrict__ w14, const float* __restrict__ w24,
                          const float* __restrict__ f0, const float* __restrict__ f1,
                          const float* __restrict__ f2,
                          float* __restrict__ out, int E) {
  extern __shared__ float smem[];
  const int tid = threadIdx.x;

  float* L11 = smem + OFF_W11;
  float* L12 = smem + OFF_W12;
  float* L13 = smem + OFF_W13;
  float* L14 = smem + OFF_W14;
  float* LF0 = smem + OFF_F0;
  float* LF1 = smem + OFF_F1;
  float* LF2 = smem + OFF_F2;

  // ---- stage all weights into LDS once per block (async, no VGPR bounce) ----
  stage_async(L11, w11, 128, 128, 136, tid);
  stage_async(L12, w12, 128,  64,  72, tid);
  stage_async(L13, w13,  64,  64,  72, tid);
  stage_async(L14, w14,  64, 128, 136, tid);
  stage_async(LF0, f0,  128,  64,  72, tid);
  stage_async(LF1, f1,   64,  64,  72, tid);
  stage_async(LF2, f2,   64, 192, 200, tid);
  asm volatile("s_wait_asynccnt 0x0" ::: "memory");
  __syncthreads();

  const int wave = tid >> 5;
  const int lane = tid & 31;
  const int n = lane & 15;   // column / M-row index within fragment
  const int h = lane >> 4;   // K-half selector
  const int e0 = (blockIdx.x * 8 + wave) * 16;
  if (e0 >= E) return;

  float* hbuf = smem + OFF_H + wave * (16 * 68);

  const float INV_S128 = 0.08838834764831845f;  // 1/sqrt(128)
  const float INV_S64  = 0.125f;                // 1/sqrt(64)
  const float C0f      = 0.07216878364870323f;  // sqrt(1/192); C1/sqrt(3) == C0
  const float INV3     = 0.5773502691896258f;   // 1/sqrt(3)

  // =====================  FC chain: h0 = silu(x@f0 /√128)  ==================
  #pragma unroll
  for (int t = 0; t < 4; ++t) {
    v8f acc = {};
    #pragma unroll
    for (int kc = 0; kc < 32; ++kc)
      acc = wmma4(glb_a(feaw, e0, 128, 0, kc, n, h),
                  lds_b(LF0 + 16 * t, 72, kc, h, n), acc);
    #pragma unroll
    for (int r = 0; r < 8; ++r)
      hbuf[(r + 8 * h) * 68 + 16 * t + n] = silu(acc[r] * INV_S128);
  }

  // h1 = silu(h0@f1 /√64): read h0 frags from LDS, then overwrite scratch
  v8f hn[4];
  #pragma unroll
  for (int t = 0; t < 4; ++t) {
    v8f acc = {};
    #pragma unroll
    for (int kc = 0; kc < 16; ++kc)
      acc = wmma4(lds_a(hbuf, kc, n, h), lds_b(LF1 + 16 * t, 72, kc, h, n), acc);
    hn[t] = acc;
  }
  #pragma unroll
  for (int t = 0; t < 4; ++t)
    #pragma unroll
    for (int r = 0; r < 8; ++r)
      hbuf[(r + 8 * h) * 68 + 16 * t + n] = silu(hn[t][r] * INV_S64);

  // =======  dot11 A-fragments: dot11[e,u] = sum_m v1[e,u,m]*v2[e,m]  ========
  const float4 qa = *reinterpret_cast<const float4*>(fea2 + 4 * (e0 + n));
  v2f dA[16];
  #pragma unroll
  for (int kc = 0; kc < 16; ++kc) {
    int u = 4 * kc + 2 * h;
    const float* p = fea1 + (size_t)(e0 + n) * 320 + 128 + 3 * u;
    dA[kc].x = p[0] * qa.y + p[1] * qa.z + p[2] * qa.w;
    dA[kc].y = p[3] * qa.y + p[4] * qa.z + p[5] * qa.w;
  }

  // ====  register-cache s1 A-fragments (reused by 12 GEMM column tiles)  ====
  v2f sA[32];
  #pragma unroll
  for (int kc = 0; kc < 32; ++kc)
    sA[kc] = glb_a(fea1, e0, 320, 0, kc, n, h);

  // ====  register-cache fea2 rows used by the epilogues (8 rows/lane)  ======
  float4 q8[8];
  #pragma unroll
  for (int r = 0; r < 8; ++r)
    q8[r] = *reinterpret_cast<const float4*>(fea2 + 4 * (e0 + r + 8 * h));

  // ====================  z0: 8 column tiles of width 16  ====================
  for (int t = 0; t < 8; ++t) {
    v8f p1 = {}, p4 = {}, wf = {};
    #pragma unroll
    for (int kc = 0; kc < 32; ++kc)
      p1 = wmma4(sA[kc], lds_b(L11 + 16 * t, 136, kc, h, n), p1);
    #pragma unroll
    for (int kc = 0; kc < 16; ++kc) {
      p4 = wmma4(dA[kc], lds_b(L14 + 16 * t, 136, kc, h, n), p4);
      wf = wmma4(lds_a(hbuf, kc, n, h), lds_b(LF2 + 16 * t, 200, kc, h, n), wf);
    }
    const float w21v = w21[16 * t + n];
    const float w24v = w24[16 * t + n];
    #pragma unroll
    for (int r = 0; r < 8; ++r) {
      int edge = e0 + r + 8 * h;
      float val = C0f * (p1[r] * w21v * q8[r].x + INV3 * p4[r] * w24v) *
                  (wf[r] * INV_S64);
      out[(size_t)edge * 320 + 16 * t + n] = val;
    }
  }

  // ====================  z1: 4 u-tiles x 3 vector comps  ====================
  for (int t = 0; t < 4; ++t) {
    v8f aT = {}, b0 = {}, b1 = {}, b2 = {}, wf = {};
    #pragma unroll
    for (int kc = 0; kc < 32; ++kc)
      aT = wmma4(sA[kc], lds_b(L12 + 16 * t, 72, kc, h, n), aT);
    #pragma unroll
    for (int kc = 0; kc < 16; ++kc) {
      int u = 4 * kc + 2 * h;
      const float* p = fea1 + (size_t)(e0 + n) * 320 + 128 + 3 * u;
      v2f a0, a1, a2;
      a0.x = p[0]; a0.y = p[3];
      a1.x = p[1]; a1.y = p[4];
      a2.x = p[2]; a2.y = p[5];
      v2f bw = lds_b(L13 + 16 * t, 72, kc, h, n);
      b0 = wmma4(a0, bw, b0);
      b1 = wmma4(a1, bw, b1);
      b2 = wmma4(a2, bw, b2);
      wf = wmma4(lds_a(hbuf, kc, n, h),
                 lds_b(LF2 + 128 + 16 * t, 200, kc, h, n), wf);
    }
    const float w22v = w22[16 * t + n];
    const float w23v = w23[16 * t + n];
    #pragma unroll
    for (int r = 0; r < 8; ++r) {
      int edge = e0 + r + 8 * h;
      float au  = aT[r] * w22v;
      float wfv = wf[r] * INV_S64;
      float cm  = w23v * q8[r].x;
      float* o = out + (size_t)edge * 320 + 128 + 3 * (16 * t + n);
      o[0] = C0f * (au * q8[r].y + b0[r] * cm) * wfv;
      o[1] = C0f * (au * q8[r].z + b1[r] * cm) * wfv;
      o[2] = C0f * (au * q8[r].w + b2[r] * cm) * wfv;
    }
  }
}

extern "C" void kernel_launch(void* const* d_in, const int* in_sizes, int n_in,
                              void* d_out, int out_size, void* d_ws, size_t ws_size,
                              hipStream_t stream) {
  const float* fea1 = (const float*)d_in[0];
  const float* fea2 = (const float*)d_in[1];
  const float* feaw = (const float*)d_in[2];
  // d_in[3] = batch_edge (unused by reference output)
  const float* w11 = (const float*)d_in[4];
  const float* w21 = (const float*)d_in[5];
  const float* w12 = (const float*)d_in[6];
  const float* w22 = (const float*)d_in[7];
  const float* w13 = (const float*)d_in[8];
  const float* w23 = (const float*)d_in[9];
  const float* w14 = (const float*)d_in[10];
  const float* w24 = (const float*)d_in[11];
  const float* f0  = (const float*)d_in[12];
  const float* f1  = (const float*)d_in[13];
  const float* f2  = (const float*)d_in[14];
  float* out = (float*)d_out;

  const int E = in_sizes[1] / 4;  // fea_in2 is (E,4)
  const size_t shmem = (size_t)LDS_FLOATS * sizeof(float);  // 301056 B < 320 KB

  (void)hipFuncSetAttribute(reinterpret_cast<const void*>(equiconv_wmma_kernel),
                            hipFuncAttributeMaxDynamicSharedMemorySize, (int)shmem);

  const int blocks = (E + 127) / 128;  // 8 waves/block x 16 edges/wave
  equiconv_wmma_kernel<<<blocks, 256, shmem, stream>>>(
      fea1, fea2, feaw, w11, w21, w12, w22, w13, w23, w14, w24,
      f0, f1, f2, out, E);
}